// res_gcn_d_46127948759543
// MI455X (gfx1250) — compile-verified
//
#include <hip/hip_runtime.h>
#include <hip/hip_bf16.h>

typedef __attribute__((ext_vector_type(2))) float v2f;
typedef __attribute__((ext_vector_type(8))) float v8f;

#define RGCN_B   4
#define RGCN_N   4096
#define RGCN_C   128
#define RGCN_K   16
#define RGCN_NB  3
#define NEG_SLOPE 0.01f

// ---------------------------------------------------------------------------
// Kernel 1: KNN indices. One workgroup (256 threads) per (batch, row).
// dist[j] = |xm|^2 + |xj|^2 - 2 xm.xj (matches reference formula).
// Select k+1 LARGEST (top_k largest=True), drop the first -> ranks 2..k+1.
// Ties resolved to the smallest index (jax top_k stability).
// ---------------------------------------------------------------------------
__global__ __launch_bounds__(256) void knn_kernel(const float* __restrict__ xyz,
                                                  int* __restrict__ idx_out,
                                                  int n) {
  __shared__ float dist[RGCN_N];
  __shared__ float redV[256];
  __shared__ int   redI[256];

  const int tid = threadIdx.x;
  const int m   = blockIdx.x;
  const int bb  = blockIdx.y;
  const float* xb = xyz + (size_t)bb * n * 3;

  const float xm0 = xb[m * 3 + 0];
  const float xm1 = xb[m * 3 + 1];
  const float xm2 = xb[m * 3 + 2];
  const float sqm = xm0 * xm0 + xm1 * xm1 + xm2 * xm2;

  for (int j = tid; j < n; j += 256) {
    const float x0 = xb[j * 3 + 0];
    const float x1 = xb[j * 3 + 1];
    const float x2 = xb[j * 3 + 2];
    const float sqj = x0 * x0 + x1 * x1 + x2 * x2;
    const float dot = xm0 * x0 + xm1 * x1 + xm2 * x2;
    dist[j] = sqm + sqj - 2.0f * dot;
  }
  __syncthreads();

  for (int r = 0; r < RGCN_K + 1; ++r) {
    float bv = -3.4e38f;
    int   bi = 0x7fffffff;
    for (int j = tid; j < n; j += 256) {
      const float v = dist[j];
      if (v > bv || (v == bv && j < bi)) { bv = v; bi = j; }
    }
    redV[tid] = bv;
    redI[tid] = bi;
    __syncthreads();
    for (int s = 128; s > 0; s >>= 1) {
      if (tid < s) {
        const float ov = redV[tid + s];
        const int   oi = redI[tid + s];
        if (ov > redV[tid] || (ov == redV[tid] && oi < redI[tid])) {
          redV[tid] = ov;
          redI[tid] = oi;
        }
      }
      __syncthreads();
    }
    const int win = redI[0];
    if (tid == 0) {
      if (r > 0) idx_out[((size_t)bb * n + m) * RGCN_K + (r - 1)] = win;
      dist[win] = -3.4e38f;
    }
    __syncthreads();
  }
}

// ---------------------------------------------------------------------------
// Kernel 2: build transposed combined weights Wt[i][kk][o], kk in [0,256):
//   kk <  128 -> Wc[i][o][kk]
//   kk >= 128 -> Wg[i][o][kk-128]
// so B-fragment loads (consecutive o across lanes) are coalesced.
// ---------------------------------------------------------------------------
__global__ __launch_bounds__(256) void prep_w(const float* __restrict__ Wc,
                                              const float* __restrict__ Wg,
                                              float* __restrict__ Wt) {
  const int g = blockIdx.x * 256 + threadIdx.x;   // 0 .. 3*256*128-1
  const int i   = g / (256 * RGCN_C);
  const int rem = g % (256 * RGCN_C);
  const int kk  = rem >> 7;
  const int o   = rem & (RGCN_C - 1);
  float v;
  if (kk < RGCN_C) v = Wc[((size_t)i * RGCN_C + o) * RGCN_C + kk];
  else             v = Wg[((size_t)i * RGCN_C + o) * RGCN_C + (kk - RGCN_C)];
  Wt[g] = v;
}

// ---------------------------------------------------------------------------
// Kernel 3: one residual block pass.
// Workgroup = 128 threads (4 wave32), tile = 32 rows x 128 cols.
// A-tile (LDS, 32 x 256, padded stride 260): [leaky(pts) | sum_k leaky(gather)]
// GEMM via V_WMMA_F32_16X16X4_F32, K accumulated in steps of 4 over 256.
// Epilogue: (acc + bc[o] + K*bg[o]) / (K+1) + shortcut.
// ---------------------------------------------------------------------------
#define TPAD 260

__global__ __launch_bounds__(128) void rgcn_block(const float* __restrict__ pin,
                                                  const float* __restrict__ Wt,
                                                  const float* __restrict__ bcv,
                                                  const float* __restrict__ bgv,
                                                  const int* __restrict__ idx,
                                                  float* __restrict__ pout,
                                                  int n) {
  __shared__ float At[32][TPAD];
  __shared__ int   idxS[32][RGCN_K];

  const int tid  = threadIdx.x;
  const int bb   = blockIdx.y;
  const int row0 = blockIdx.x * 32;
  const size_t base = (size_t)bb * n * RGCN_C;

  // stage neighbor indices for the 32 rows
  for (int e = tid; e < 32 * RGCN_K; e += 128) {
    const int r = e >> 4, j = e & (RGCN_K - 1);
    idxS[r][j] = idx[((size_t)bb * n + row0 + r) * RGCN_K + j];
  }
  __syncthreads();

  // phase 1: leaky(points) rows  (columns 0..127)
  for (int e = tid; e < 32 * RGCN_C; e += 128) {
    const int r = e >> 7, c = e & (RGCN_C - 1);
    const float x = pin[base + (size_t)(row0 + r) * RGCN_C + c];
    At[r][c] = (x >= 0.f) ? x : NEG_SLOPE * x;
  }
  // phase 2: gathered neighbor sums (columns 128..255)
  for (int e = tid; e < 32 * RGCN_C; e += 128) {
    const int r = e >> 7, c = e & (RGCN_C - 1);
    float s = 0.f;
#pragma unroll
    for (int j = 0; j < RGCN_K; ++j) {
      const float x = pin[base + (size_t)idxS[r][j] * RGCN_C + c];
      s += (x >= 0.f) ? x : NEG_SLOPE * x;
    }
    At[r][RGCN_C + c] = s;
  }
  __syncthreads();

  // WMMA phase: wave w -> rows [(w&1)*16, +16), cols [(w>>1)*64, +64)
  const int lane  = tid & 31;
  const int wave  = tid >> 5;
  const int rBase = (wave & 1) * 16;
  const int cBase = (wave >> 1) * 64;
  const int mrow  = rBase + (lane & 15);
  const int khalf = (lane >> 4) << 1;   // lanes 0-15 -> K+0/K+1, 16-31 -> K+2/K+3
  const int ncol  = lane & 15;

  v8f acc[4] = {v8f{0.f}, v8f{0.f}, v8f{0.f}, v8f{0.f}};

  for (int kk = 0; kk < 256; kk += 4) {
    v2f a;
    a.x = At[mrow][kk + khalf + 0];
    a.y = At[mrow][kk + khalf + 1];
#pragma unroll
    for (int t = 0; t < 4; ++t) {
      const int o = cBase + t * 16 + ncol;
      v2f bf;
      bf.x = Wt[(size_t)(kk + khalf + 0) * RGCN_C + o];
      bf.y = Wt[(size_t)(kk + khalf + 1) * RGCN_C + o];
      acc[t] = __builtin_amdgcn_wmma_f32_16x16x4_f32(
          false, a, false, bf, (short)0, acc[t], false, false);
    }
  }

  // epilogue: C-layout lanes 0-15 -> M=r, lanes 16-31 -> M=8+r
  const int mAdd = (lane >> 4) << 3;
#pragma unroll
  for (int t = 0; t < 4; ++t) {
    const int o = cBase + t * 16 + ncol;
    const float bias = bcv[o] + (float)RGCN_K * bgv[o];
#pragma unroll
    for (int r = 0; r < 8; ++r) {
      const int mr = rBase + r + mAdd;
      const size_t gi = base + (size_t)(row0 + mr) * RGCN_C + o;
      pout[gi] = (acc[t][r] + bias) / (float)(RGCN_K + 1) + pin[gi];
    }
  }
}

// ---------------------------------------------------------------------------
extern "C" void kernel_launch(void* const* d_in, const int* in_sizes, int n_in,
                              void* d_out, int out_size, void* d_ws, size_t ws_size,
                              hipStream_t stream) {
  (void)in_sizes; (void)n_in; (void)out_size; (void)ws_size;
  const float* xyz    = (const float*)d_in[0];
  const float* points = (const float*)d_in[1];
  // d_in[2]=k, d_in[3]=n_cout, d_in[4]=n_blocks (device scalars; fixed by setup)
  const float* Wc = (const float*)d_in[5];
  const float* bc = (const float*)d_in[6];
  const float* Wg = (const float*)d_in[7];
  const float* bg = (const float*)d_in[8];

  char* ws = (char*)d_ws;
  int*   d_idx = (int*)ws;                                   // 4*4096*16*4 = 1 MB
  float* d_wt  = (float*)(ws + (1u << 20));                  // 3*256*128*4 = 384 KB
  float* d_pa  = (float*)(ws + (1u << 20) + (512u << 10));   // 8.4 MB ping
  float* d_pb  = (float*)(ws + (1u << 20) + (512u << 10) + (9u << 20)); // pong

  // 1) KNN indices (computed once, reused by all 3 blocks)
  knn_kernel<<<dim3(RGCN_N, RGCN_B), 256, 0, stream>>>(xyz, d_idx, RGCN_N);

  // 2) transpose/combine weights: Wt[i][k][o], k in [0,256)
  prep_w<<<(RGCN_NB * 256 * RGCN_C) / 256, 256, 0, stream>>>(Wc, Wg, d_wt);

  // 3) three residual block passes, ping-ponging through workspace
  const float* src[RGCN_NB] = {points, d_pa, d_pb};
  float*       dst[RGCN_NB] = {d_pa, d_pb, (float*)d_out};
  for (int i = 0; i < RGCN_NB; ++i) {
    rgcn_block<<<dim3(RGCN_N / 32, RGCN_B), 128, 0, stream>>>(
        src[i], d_wt + (size_t)i * 256 * RGCN_C,
        bc + i * RGCN_C, bg + i * RGCN_C, d_idx, dst[i], RGCN_N);
  }
}